// LlamaAttention_49478023250125
// MI455X (gfx1250) — compile-verified
//
#include <hip/hip_runtime.h>
#include <math.h>

#define H      4096
#define NH     32
#define NKV    8
#define HD     128
#define CACHE  2048
#define GROUPS 4                       // NH / NKV
#define SCALE  0.08838834764831845f    // 1/sqrt(128)
#define SCPAD  1056                    // padded score columns (>= 1040, even)

typedef __attribute__((ext_vector_type(2))) float v2f;
typedef __attribute__((ext_vector_type(8))) float v8f;

__device__ __forceinline__ float wave_sum(float v) {
#pragma unroll
    for (int o = 16; o > 0; o >>= 1) v += __shfl_xor(v, o, 32);
    return v;
}
__device__ __forceinline__ float wave_max(float v) {
#pragma unroll
    for (int o = 16; o > 0; o >>= 1) v = fmaxf(v, __shfl_xor(v, o, 32));
    return v;
}

// ---------------------------------------------------------------------------
// Kernel 1: fused Q/K/V projection GEMV. One wave per output row.
// rows [0,4096) -> q ; [4096,5120) -> k ; [5120,6144) -> v
// ---------------------------------------------------------------------------
__global__ __launch_bounds__(256) void qkv_gemv(const float* __restrict__ x,
                                                const float* __restrict__ qw,
                                                const float* __restrict__ kw,
                                                const float* __restrict__ vw,
                                                float* __restrict__ out) {
    const int wave = threadIdx.x >> 5;
    const int lane = threadIdx.x & 31;
    const int row  = blockIdx.x * 8 + wave;           // 768 blocks * 8 waves
    const float* wrow;
    if (row < H)                 wrow = qw + (size_t)row * H;
    else if (row < H + NKV * HD) wrow = kw + (size_t)(row - H) * H;
    else                         wrow = vw + (size_t)(row - H - NKV * HD) * H;

    const float4* w4 = (const float4*)wrow;
    const float4* x4 = (const float4*)x;
    float acc = 0.f;
#pragma unroll 4
    for (int i = lane; i < H / 4; i += 32) {
        float4 a = w4[i];
        float4 b = x4[i];
        acc += a.x * b.x + a.y * b.y + a.z * b.z + a.w * b.w;
    }
    acc = wave_sum(acc);
    if (lane == 0) out[row] = acc;
}

// ---------------------------------------------------------------------------
// Kernel 2: rotate-half RoPE in place on q (heads 0..31) and new k (32..39).
// ---------------------------------------------------------------------------
__global__ __launch_bounds__(256) void rope_qk(float* __restrict__ qkv,
                                               const float* __restrict__ cosc,
                                               const float* __restrict__ sinc,
                                               const int* __restrict__ cur_pos) {
    const int pos = *cur_pos;
    const int tid = blockIdx.x * blockDim.x + threadIdx.x;   // 2560 pairs
    if (tid >= (NH + NKV) * (HD / 2)) return;
    const int head = tid / (HD / 2);
    const int d    = tid % (HD / 2);
    float* base = qkv + head * HD;
    const float c = cosc[(size_t)pos * HD + d];
    const float s = sinc[(size_t)pos * HD + d];
    const float x1 = base[d];
    const float x2 = base[d + HD / 2];
    base[d]          = x1 * c - x2 * s;
    base[d + HD / 2] = x2 * c + x1 * s;
}

// ---------------------------------------------------------------------------
// Kernel 3: attention core. One workgroup (8 waves) per KV head.
// S = Qg(4x128 padded to 16) x K^T   via V_WMMA_F32_16X16X4_F32
// softmax per head (wave-local), then ctx = P x V via WMMA.
// LDS operands are padded with zero rows so A-fragment loads are
// unconditional ds_load_b64 (no exec-divergent loads).
// ISA 7.12.2 f32 layouts: A/B lane L comp j -> K = j + 2*(L>>4), row/col = L&15
//                         D  VGPR r, lane L -> row = r + 8*(L>>4), col = L&15
// ---------------------------------------------------------------------------
__global__ __launch_bounds__(256) void attn_wmma(const float* __restrict__ qkv,
                                                 const float* __restrict__ kcache,
                                                 const float* __restrict__ vcache,
                                                 const float* __restrict__ mask,
                                                 const int* __restrict__ cur_pos,
                                                 float* __restrict__ ctx) {
    const int kv   = blockIdx.x;          // 0..7
    const int tid  = threadIdx.x;
    const int wave = tid >> 5;
    const int lane = tid & 31;
    const int pos    = *cur_pos;
    const int ntiles = (pos + 16) >> 4;   // ceil((pos+1)/16) = 65 for pos=1024
    const int npos   = ntiles * 16;       // 1040

    __shared__ __align__(16) float qs[16 * HD];          // rows 4..15 zero
    __shared__ __align__(16) float sc[GROUPS + 1][SCPAD]; // row 4 = zero dummy

    const float* qbase = qkv + kv * GROUPS * HD;
    for (int i = tid; i < 16 * HD; i += 256)
        qs[i] = (i < GROUPS * HD) ? qbase[i] : 0.f;
    for (int i = tid; i < SCPAD; i += 256)
        sc[GROUPS][i] = 0.f;
    __syncthreads();

    const int m     = lane & 15;          // A row / B col / D col index
    const int khalf = (lane >> 4) * 2;    // K sub-index pair for this half-wave

    // ---- scores: S[m][p] = sum_k Q[m][k] * K[p][k] -------------------------
    const float* knew  = qkv + H + kv * HD;
    const float* krows = kcache + (size_t)kv * CACHE * HD;
    for (int T = wave; T < ntiles; T += 8) {
        const int p = T * 16 + m;         // this lane's K-cache row (B column)
        const float* rp = (p == pos) ? knew : (krows + (size_t)p * HD);
        v8f c = {};
#pragma unroll 8
        for (int kb = 0; kb < HD; kb += 4) {
            const int ks = kb + khalf;
            v2f a = *(const v2f*)&qs[m * HD + ks];     // ds_load_b64
            v2f b = *(const v2f*)&rp[ks];              // global_load_b64
            c = __builtin_amdgcn_wmma_f32_16x16x4_f32(false, a, false, b,
                                                      (short)0, c, false, false);
        }
        if (lane < 16) {
            const int   pcol = T * 16 + lane;
            const float mk   = mask[pcol];
#pragma unroll
            for (int r = 0; r < GROUPS; ++r)
                sc[r][pcol] = c[r] * SCALE + mk;
        }
    }
    __syncthreads();

    // ---- softmax: one wave per grouped head --------------------------------
    if (wave < GROUPS) {
        const int g = wave;
        float mx = -3.0e38f;
        for (int p = lane; p < npos; p += 32) mx = fmaxf(mx, sc[g][p]);
        mx = wave_max(mx);
        float sum = 0.f;
        for (int p = lane; p < npos; p += 32) {
            const float e = expf(sc[g][p] - mx);
            sc[g][p] = e;
            sum += e;
        }
        sum = wave_sum(sum);
        const float inv = 1.0f / sum;
        for (int p = lane; p < npos; p += 32) sc[g][p] *= inv;
    }
    __syncthreads();

    // ---- context: ctx[m][n] = sum_p P[m][p] * V[p][n] ----------------------
    const float* vnew  = qkv + H + NKV * HD + kv * HD;
    const float* vrows = vcache + (size_t)kv * CACHE * HD;
    const int ridx = (m < GROUPS) ? m : GROUPS;        // zero dummy row for m>=4
    for (int nt = wave; nt < HD / 16; nt += 8) {       // 8 n-tiles, 1 per wave
        const int n = nt * 16 + m;                     // B column (head dim)
        v8f c = {};
        for (int pb = 0; pb < npos; pb += 4) {
            const int p0 = pb + khalf;
            v2f a = *(const v2f*)&sc[ridx][p0];        // ds_load_b64
            v2f b;
            b[0] = (p0 > pos)      ? 0.f
                 : (p0 == pos)     ? vnew[n] : vrows[(size_t)p0 * HD + n];
            b[1] = (p0 + 1 > pos)  ? 0.f
                 : (p0 + 1 == pos) ? vnew[n] : vrows[(size_t)(p0 + 1) * HD + n];
            c = __builtin_amdgcn_wmma_f32_16x16x4_f32(false, a, false, b,
                                                      (short)0, c, false, false);
        }
        if (lane < 16) {
#pragma unroll
            for (int r = 0; r < GROUPS; ++r)
                ctx[(size_t)(kv * GROUPS + r) * HD + nt * 16 + lane] = c[r];
        }
    }
}

// ---------------------------------------------------------------------------
// Kernel 4: output projection GEMV, out[o] = sum_d ctx[d] * o_w[o][d].
// ---------------------------------------------------------------------------
__global__ __launch_bounds__(256) void out_gemv(const float* __restrict__ ctx,
                                                const float* __restrict__ ow,
                                                float* __restrict__ out) {
    const int wave = threadIdx.x >> 5;
    const int lane = threadIdx.x & 31;
    const int row  = blockIdx.x * 8 + wave;           // 512 blocks
    const float4* w4 = (const float4*)(ow + (size_t)row * H);
    const float4* x4 = (const float4*)ctx;
    float acc = 0.f;
#pragma unroll 4
    for (int i = lane; i < H / 4; i += 32) {
        float4 a = w4[i];
        float4 b = x4[i];
        acc += a.x * b.x + a.y * b.y + a.z * b.z + a.w * b.w;
    }
    acc = wave_sum(acc);
    if (lane == 0) out[row] = acc;
}

extern "C" void kernel_launch(void* const* d_in, const int* in_sizes, int n_in,
                              void* d_out, int out_size, void* d_ws, size_t ws_size,
                              hipStream_t stream) {
    (void)in_sizes; (void)n_in; (void)out_size; (void)ws_size;
    const float* x      = (const float*)d_in[0];   // hidden_states (1,1,4096)
    const float* mask   = (const float*)d_in[1];   // attention_mask (2048)
    const float* kcache = (const float*)d_in[2];   // (1,8,2048,128)
    const float* vcache = (const float*)d_in[3];
    const float* qw     = (const float*)d_in[4];   // (4096,4096)
    const float* kw     = (const float*)d_in[5];   // (1024,4096)
    const float* vw     = (const float*)d_in[6];   // (1024,4096)
    const float* ow     = (const float*)d_in[7];   // (4096,4096)
    const float* cosc   = (const float*)d_in[8];   // (1,4096,128)
    const float* sinc   = (const float*)d_in[9];
    const int*   cpos   = (const int*)d_in[11];    // current_pos

    float* ws  = (float*)d_ws;
    float* qkv = ws;            // q[4096] | k_new[1024] | v_new[1024]
    float* ctx = ws + 6144;     // ctx[4096]

    qkv_gemv<<<(H + 2 * NKV * HD) / 8, 256, 0, stream>>>(x, qw, kw, vw, qkv);
    rope_qk<<<10, 256, 0, stream>>>(qkv, cosc, sinc, cpos);
    attn_wmma<<<NKV, 256, 0, stream>>>(qkv, kcache, vcache, mask, cpos, ctx);
    out_gemv<<<H / 8, 256, 0, stream>>>(ctx, ow, (float*)d_out);
}